// INGP_36790689857891
// MI455X (gfx1250) — compile-verified
//
#include <hip/hip_runtime.h>
#include <cmath>

// ---------------- Instant-NGP hashgrid config (matches reference) ----------
constexpr int      kLevels = 16;
constexpr int      kFeat   = 2;
constexpr unsigned kT      = 1u << 19;
constexpr unsigned kMask   = kT - 1u;
constexpr unsigned kP1     = 2654435761u;   // NGP spatial-hash primes
constexpr unsigned kP2     = 805459861u;    // (prime for dim0 is 1)

// native clang vector: valid for __builtin_nontemporal_* and 8-byte loads
typedef float f32x2 __attribute__((ext_vector_type(2)));

struct ResTable { float r[kLevels]; };      // passed by value -> kernarg SGPRs

// One thread per point; 16 levels fully unrolled; 8 f32x2 gathers per level.
// Tables (64 MB total) stay resident in the 192 MB L2; the 64 MB output
// stream and the one-shot point reads use non-temporal hints so they do not
// evict the gather working set.
__global__ __launch_bounds__(256) void ingp_hash_encode(
    const float* __restrict__ points,   // [N,3]
    const float* __restrict__ tables,   // [L,T,F] f32
    float*       __restrict__ out,      // [N, L*F] f32
    ResTable res, int n)
{
    const int i = blockIdx.x * 256 + threadIdx.x;
    if (i >= n) return;

    const float px = __builtin_nontemporal_load(points + 3 * i + 0);
    const float py = __builtin_nontemporal_load(points + 3 * i + 1);
    const float pz = __builtin_nontemporal_load(points + 3 * i + 2);

    // normalize [-1,1] -> [0,1]  (exact: add 1, multiply by 0.5)
    const float x = (px + 1.0f) * 0.5f;
    const float y = (py + 1.0f) * 0.5f;
    const float z = (pz + 1.0f) * 0.5f;

    float* orow = out + (size_t)i * (kLevels * kFeat);

    #pragma unroll
    for (int l = 0; l < kLevels; ++l) {
        const float r = res.r[l];                     // SGPR constant per level
        const float fx = x * r, fy = y * r, fz = z * r;
        const float x0f = floorf(fx), y0f = floorf(fy), z0f = floorf(fz);
        const float wx = fx - x0f, wy = fy - y0f, wz = fz - z0f;

        const unsigned x0 = (unsigned)x0f;
        const unsigned y0 = (unsigned)y0f;
        const unsigned z0 = (unsigned)z0f;
        // hash components for the two planes of each dim (u32 wraparound mults)
        const unsigned hx0 = x0,            hx1 = x0 + 1u;
        const unsigned hy0 = y0 * kP1,      hy1 = hy0 + kP1;
        const unsigned hz0 = z0 * kP2,      hz1 = hz0 + kP2;

        const f32x2* tb = (const f32x2*)tables + (size_t)l * kT;

        // 8 independent gathers (global_load_b64) -> deep MLP per wave
        const f32x2 v000 = tb[(hx0 ^ hy0 ^ hz0) & kMask];
        const f32x2 v001 = tb[(hx0 ^ hy0 ^ hz1) & kMask];
        const f32x2 v010 = tb[(hx0 ^ hy1 ^ hz0) & kMask];
        const f32x2 v011 = tb[(hx0 ^ hy1 ^ hz1) & kMask];
        const f32x2 v100 = tb[(hx1 ^ hy0 ^ hz0) & kMask];
        const f32x2 v101 = tb[(hx1 ^ hy0 ^ hz1) & kMask];
        const f32x2 v110 = tb[(hx1 ^ hy1 ^ hz0) & kMask];
        const f32x2 v111 = tb[(hx1 ^ hy1 ^ hz1) & kMask];

        // trilinear blend
        const float ux = 1.0f - wx, uy = 1.0f - wy, uz = 1.0f - wz;
        const float w00 = uy * uz, w01 = uy * wz, w10 = wy * uz, w11 = wy * wz;

        f32x2 acc;
        acc.x = ux * (w00 * v000.x + w01 * v001.x + w10 * v010.x + w11 * v011.x)
              + wx * (w00 * v100.x + w01 * v101.x + w10 * v110.x + w11 * v111.x);
        acc.y = ux * (w00 * v000.y + w01 * v001.y + w10 * v010.y + w11 * v011.y)
              + wx * (w00 * v100.y + w01 * v101.y + w10 * v110.y + w11 * v111.y);

        // output is write-once streaming: non-temporal so the 64 MB stream
        // does not evict the 64 MB of hash tables resident in L2
        __builtin_nontemporal_store(acc, (f32x2*)(orow + 2 * l));
    }
}

extern "C" void kernel_launch(void* const* d_in, const int* in_sizes, int n_in,
                              void* d_out, int out_size, void* d_ws, size_t ws_size,
                              hipStream_t stream) {
    const float* points = (const float*)d_in[0];   // [N,3] f32
    const float* tables = (const float*)d_in[1];   // [16, 2^19, 2] f32
    float* out = (float*)d_out;                    // [N,32] f32
    const int n = in_sizes[0] / 3;

    // Replicate numpy's float64 RES computation exactly (same libm sequence):
    // GROWTH = exp((log(2048)-log(16))/15); RES[l] = floor(16*GROWTH**l).
    // Several levels land ~1e-2 (and l=15 ~1e-11) from integer boundaries,
    // so this must be bit-matched in double, not approximated.
    ResTable res;
    const double growth = std::exp((std::log(2048.0) - std::log(16.0)) / 15.0);
    for (int l = 0; l < kLevels; ++l) {
        res.r[l] = (float)std::floor(16.0 * std::pow(growth, (double)l));
    }

    const int grid = (n + 255) / 256;
    ingp_hash_encode<<<grid, 256, 0, stream>>>(points, tables, out, res, n);
}